// HEMLoss_41764261986862
// MI455X (gfx1250) — compile-verified
//
#include <hip/hip_runtime.h>
#include <hip/hip_bf16.h>
#include <cstdint>

// Problem geometry (fixed by reference): x,y are (32, 3, 512, 512) f32.
#define NB   32
#define NC   3
#define HW   262144            // 512*512
#define BSTR (NC * HW)         // per-batch stride in x/y
#define HARD_K 131072u         // rank (1-indexed k-th smallest) = hw - hard_ind
#define RAND_IND 26214u        // int(0.1 * hw)

typedef float v2f __attribute__((ext_vector_type(2)));
typedef float v8f __attribute__((ext_vector_type(8)));

// ---------------------------------------------------------------------------
// Pass 1: res[b,p] = sum_c |x-y|, staged through LDS with async global->LDS
// loads; also builds level-0 histogram (top 12 bits of float pattern).
// Grid: (256, 32) blocks of 256 threads; 1024 pixels per block.
// ---------------------------------------------------------------------------
__global__ __launch_bounds__(256)
void k_res_hist(const float* __restrict__ x, const float* __restrict__ y,
                float* __restrict__ res, unsigned* __restrict__ hist0) {
    __shared__ float xs[NC][1024];
    __shared__ float ys[NC][1024];
    __shared__ unsigned lh[4096];

    const int t   = threadIdx.x;
    const int b   = blockIdx.y;
    const int p0  = blockIdx.x * 1024;

    for (int i = t; i < 4096; i += 256) lh[i] = 0;

    const float* xb = x + (size_t)b * BSTR + p0;
    const float* yb = y + (size_t)b * BSTR + p0;

    // Async-stage 3 channels x 1024 floats of both tensors into LDS.
    // Each lane issues one b128 (16B) per channel per tensor: 6 async ops.
    #pragma unroll
    for (int c = 0; c < NC; ++c) {
        const int idx = t * 4;                       // 256 lanes * 4 floats = 1024
        const float* gx = xb + c * HW + idx;
        const float* gy = yb + c * HW + idx;
        unsigned lx = (unsigned)(uintptr_t)&xs[c][idx];
        unsigned ly = (unsigned)(uintptr_t)&ys[c][idx];
        asm volatile("global_load_async_to_lds_b128 %0, %1, off"
                     :: "v"(lx), "v"(gx) : "memory");
        asm volatile("global_load_async_to_lds_b128 %0, %1, off"
                     :: "v"(ly), "v"(gy) : "memory");
    }
    asm volatile("s_wait_asynccnt 0" ::: "memory");
    __syncthreads();

    float* rb = res + (size_t)b * HW + p0;
    #pragma unroll
    for (int i = 0; i < 4; ++i) {
        const int p = t + i * 256;
        float r = fabsf(xs[0][p] - ys[0][p])
                + fabsf(xs[1][p] - ys[1][p])
                + fabsf(xs[2][p] - ys[2][p]);
        rb[p] = r;
        atomicAdd(&lh[__float_as_uint(r) >> 20], 1u);
    }
    __syncthreads();

    unsigned* gh = hist0 + b * 4096;
    for (int i = t; i < 4096; i += 256)
        if (lh[i]) atomicAdd(&gh[i], lh[i]);
}

// ---------------------------------------------------------------------------
// Radix-select refinement histogram: count elements matching current prefix,
// binned on the next digit. Grid: (128, 32) x 256 threads, 2048 px/block.
// ---------------------------------------------------------------------------
__global__ __launch_bounds__(256)
void k_refine(const float* __restrict__ res, unsigned* __restrict__ hist,
              const unsigned* __restrict__ pfx,
              unsigned prefmask, int shift, unsigned binmask) {
    __shared__ unsigned lh[4096];
    const int t = threadIdx.x;
    const int b = blockIdx.y;
    for (int i = t; i < 4096; i += 256) lh[i] = 0;
    __syncthreads();

    const unsigned pv = pfx[b];
    const float* rb = res + (size_t)b * HW + blockIdx.x * 2048;
    #pragma unroll
    for (int i = 0; i < 8; ++i) {
        unsigned bits = __float_as_uint(rb[t + i * 256]);
        if ((bits & prefmask) == pv)
            atomicAdd(&lh[(bits >> shift) & binmask], 1u);
    }
    __syncthreads();
    unsigned* gh = hist + b * 4096;
    for (int i = t; i < 4096; i += 256)
        if (lh[i]) atomicAdd(&gh[i], lh[i]);
}

// ---------------------------------------------------------------------------
// Per-batch digit selection (tiny, one block per batch).
// ---------------------------------------------------------------------------
__global__ void k_select(const unsigned* __restrict__ hist,
                         unsigned* __restrict__ pfx, unsigned* __restrict__ kth,
                         float* __restrict__ thre, int shift, int nbins, int fin) {
    const int b = blockIdx.x;
    if (threadIdx.x != 0) return;
    const unsigned* h = hist + b * 4096;
    unsigned k = kth[b], cum = 0;
    int bin = nbins - 1;
    for (int i = 0; i < nbins; ++i) {
        unsigned c = h[i];
        if (cum + c >= k) { bin = i; break; }
        cum += c;
    }
    unsigned np = pfx[b] | ((unsigned)bin << shift);
    pfx[b] = np;
    kth[b] = k - cum;
    if (fin) thre[b] = __uint_as_float(np);
}

__global__ void k_init(unsigned* pfx, unsigned* kth) {
    int t = threadIdx.x;
    if (t < NB) { pfx[t] = 0u; kth[t] = HARD_K; }
}

// ---------------------------------------------------------------------------
// Masked sum of res: mask = (res > thre[b]) | ~10% hash-random.
// Wave reduction via V_WMMA_F32_16X16X4_F32 (B = ones -> row sums).
// Grid: 1024 blocks x 256 threads x 32 elements = 8,388,608 = NB*HW.
// ---------------------------------------------------------------------------
__global__ __launch_bounds__(256)
void k_loss(const float* __restrict__ res, const float* __restrict__ thre,
            float* __restrict__ partials) {
    const int t = threadIdx.x;
    const int base = blockIdx.x * (256 * 32);
    float lacc = 0.0f;
    #pragma unroll 4
    for (int i = 0; i < 32; ++i) {
        const int g = base + i * 256 + t;
        const int b = g >> 18;                 // / HW
        const unsigned p = (unsigned)g & 0x3FFFFu;
        const float r = res[g];
        const bool hard = r > thre[b];
        unsigned h = p ^ ((unsigned)b * 0x9E3779B9u) ^ 42u;
        h ^= h >> 16; h *= 0x7feb352dU; h ^= h >> 15; h *= 0x846ca68bU; h ^= h >> 16;
        const bool rnd = (h & 0x3FFFFu) < RAND_IND;
        if (hard | rnd) lacc += r;
    }

    // Wave32 reduction via WMMA: A[m][k] holds lane partials (k=0 for lanes
    // 0-15, k=2 for lanes 16-31 per the 16x4 f32 A layout); B = all-ones, so
    // D[i][j] = lacc_i + lacc_{i+16}. Lane halves then hold disjoint row sets.
    v2f a; a.x = lacc; a.y = 0.0f;
    v2f bm; bm.x = 1.0f; bm.y = 1.0f;
    v8f c = {};
    c = __builtin_amdgcn_wmma_f32_16x16x4_f32(false, a, false, bm,
                                              (short)0, c, false, false);
    float s = c[0] + c[1] + c[2] + c[3] + c[4] + c[5] + c[6] + c[7];
    float wave_sum = s + __shfl_xor(s, 16, 32);   // low half + high half rows

    __shared__ float wsum[8];
    const int wave = t >> 5, lane = t & 31;
    if (lane == 0) wsum[wave] = wave_sum;
    __syncthreads();
    if (t == 0) {
        float bs = 0.0f;
        #pragma unroll
        for (int i = 0; i < 8; ++i) bs += wsum[i];
        partials[blockIdx.x] = bs;
    }
}

__global__ void k_final(const float* __restrict__ partials, float* __restrict__ out) {
    __shared__ float sm[256];
    const int t = threadIdx.x;
    float s = 0.0f;
    for (int i = t; i < 1024; i += 256) s += partials[i];
    sm[t] = s;
    __syncthreads();
    for (int w = 128; w > 0; w >>= 1) {
        if (t < w) sm[t] += sm[t + w];
        __syncthreads();
    }
    if (t == 0) out[0] = sm[0] * (1.0f / ((float)NB * NC * 512.0f * 512.0f));
}

// ---------------------------------------------------------------------------
extern "C" void kernel_launch(void* const* d_in, const int* in_sizes, int n_in,
                              void* d_out, int out_size, void* d_ws, size_t ws_size,
                              hipStream_t stream) {
    const float* x = (const float*)d_in[0];
    const float* y = (const float*)d_in[1];
    float* out = (float*)d_out;

    // Workspace carve-up (~35.1 MB total).
    unsigned char* w = (unsigned char*)d_ws;
    float*    res   = (float*)w;                                   // NB*HW f32
    unsigned* hist  = (unsigned*)(w + (size_t)NB * HW * 4);        // 3 * NB*4096
    unsigned* hist0 = hist;
    unsigned* hist1 = hist + NB * 4096;
    unsigned* hist2 = hist + 2 * NB * 4096;
    unsigned* pfx   = hist + 3 * NB * 4096;                        // NB
    unsigned* kth   = pfx + NB;                                    // NB
    float*    thre  = (float*)(kth + NB);                          // NB
    float*    parts = thre + NB;                                   // 1024

    hipMemsetAsync(hist, 0, (size_t)3 * NB * 4096 * sizeof(unsigned), stream);
    k_init<<<1, 32, 0, stream>>>(pfx, kth);

    k_res_hist<<<dim3(256, NB), 256, 0, stream>>>(x, y, res, hist0);
    k_select<<<NB, 32, 0, stream>>>(hist0, pfx, kth, thre, 20, 4096, 0);
    k_refine<<<dim3(128, NB), 256, 0, stream>>>(res, hist1, pfx, 0xFFF00000u, 8, 0xFFFu);
    k_select<<<NB, 32, 0, stream>>>(hist1, pfx, kth, thre, 8, 4096, 0);
    k_refine<<<dim3(128, NB), 256, 0, stream>>>(res, hist2, pfx, 0xFFFFFF00u, 0, 0xFFu);
    k_select<<<NB, 32, 0, stream>>>(hist2, pfx, kth, thre, 0, 256, 1);

    k_loss<<<1024, 256, 0, stream>>>(res, thre, parts);
    k_final<<<1, 256, 0, stream>>>(parts, out);
}